// SpatialRadiusEdge_37495064494462
// MI455X (gfx1250) — compile-verified
//
#include <hip/hip_runtime.h>
#include <hip/hip_bf16.h>

// adj[b,i,j] = 1.0 iff  T[b] <= j < T[b]+taus[b]  &&  i < j  &&  ||p_i - p_j|| < 0.25
// p = nodes[b, :, 0:2].  Output 8x2048x2048 f32 = 134 MB -> pure store-BW bound on MI455X
// (23.3 TB/s => ~5.8us floor).  dist^2 computed as a rank-4 matmul via V_WMMA_F32_16X16X4_F32:
//   A row m  = [x_i, y_i, x_i^2+y_i^2, 1]
//   B col n  = [-2x_j, -2y_j, 1, x_j^2+y_j^2]
//   (A*B)[m][n] = dist^2(i0+m, j0+n)
// One wave32 per 16x16 output tile; branchless masks; NT b32 stores with constant
// row offsets (ioffset = r*8192B) so each tile is 1 addr setup + 8 coalesced stores.

typedef __attribute__((ext_vector_type(2))) float v2f;
typedef __attribute__((ext_vector_type(8))) float v8f;

#define NN     2048
#define NBATCH 8
#define FEAT   16
#define TILES_PER_DIM   (NN / 16)                       // 128
#define TILES_PER_BATCH (TILES_PER_DIM * TILES_PER_DIM) // 16384
#define WAVES_PER_BLOCK 8

__global__ __launch_bounds__(256) void SpatialRadiusEdge_kernel(
    const float* __restrict__ nodes,   // [8, 2048, 16] f32
    const int*   __restrict__ T,       // [8] i32
    const int*   __restrict__ taus,    // [8] i32
    float*       __restrict__ out)     // [8, 2048, 2048] f32
{
    const int lane = threadIdx.x & 31;
    const int wave = threadIdx.x >> 5;          // wave32: 8 waves per 256-thread block

    // linear tile id -> (batch, tile-row, tile-col)
    const int t   = blockIdx.x * WAVES_PER_BLOCK + wave;
    const int b   = t >> 14;                    // / TILES_PER_BATCH
    const int rem = t & (TILES_PER_BATCH - 1);
    const int i0  = (rem >> 7) << 4;            // tile row * 16
    const int j0  = (rem & (TILES_PER_DIM - 1)) << 4;

    const int  l16 = lane & 15;
    const bool lo  = lane < 16;                 // low half-wave holds K=0,1; high holds K=2,3

    const float* nb = nodes + (size_t)b * NN * FEAT;

    // Unconditional scalar-ish mask loads (wave-uniform address) -- issue early,
    // they overlap the position loads and the WMMA.
    const int Tb = T[b];
    const int tb = taus[b];

    // Row (A-side) and column (B-side) 2D positions; features 0..1 contiguous -> b64 loads.
    const int irow = i0 + l16;
    const int jcol = j0 + l16;
    const v2f pi = *(const v2f*)(nb + (size_t)irow * FEAT);
    const v2f pj = *(const v2f*)(nb + (size_t)jcol * FEAT);

    const float normi = pi.x * pi.x + pi.y * pi.y;
    const float normj = pj.x * pj.x + pj.y * pj.y;

    // A (16x4 f32, 2 VGPRs): lanes 0-15 -> K=0,1 ; lanes 16-31 -> K=2,3   (ISA 7.12.2)
    v2f A;
    A.x = lo ? pi.x : normi;    // K=0 : x_i          | K=2 : |p_i|^2
    A.y = lo ? pi.y : 1.0f;     // K=1 : y_i          | K=3 : 1
    // B (4x16 f32, 2 VGPRs): lanes hold N=0..15; low half K=0,1, high half K=2,3
    v2f Bm;
    Bm.x = lo ? -2.0f * pj.x : 1.0f;    // K=0 : -2x_j  | K=2 : 1
    Bm.y = lo ? -2.0f * pj.y : normj;   // K=1 : -2y_j  | K=3 : |p_j|^2

    // D = A x B + 0 : full 16x16 tile of squared distances in one matrix op.
    v8f c = {};
    c = __builtin_amdgcn_wmma_f32_16x16x4_f32(
        /*neg_a=*/false, A, /*neg_b=*/false, Bm,
        /*c_mod=*/(short)0, c, /*reuse_a=*/false, /*reuse_b=*/false);

    // Branchless column-window mask: j in [T, T+tau)  <=>  (unsigned)(j-T) < tau.
    const bool jnew = ((unsigned)(jcol - Tb)) < (unsigned)tb;

    // Causal i<j is monotone in r: row ir = i0 + mbase + r  ->  causal iff r < jr.
    // Fold jnew in by clamping jr to 0 when the column is outside the window.
    const int mbase = lo ? 0 : 8;
    const int jr    = jnew ? (jcol - i0 - mbase) : 0;

    const float R2 = 0.25f * 0.25f;
    // C/D layout: lane L, VGPR r -> row m = mbase + r, col n = L%16.
    // Single per-lane base pointer; row step is the compile-time constant NN
    // -> global_store_b32 with ioffset = r*8192 bytes, lanes 0-15 cover 64B rows.
    float* prow = out + (((size_t)b * NN + (size_t)(i0 + mbase)) * NN + (size_t)(j0 + l16));
#pragma unroll
    for (int r = 0; r < 8; ++r) {
        const float v = ((r < jr) & (c[r] < R2)) ? 1.0f : 0.0f;
        __builtin_nontemporal_store(v, prow + (size_t)r * NN);
    }
}

extern "C" void kernel_launch(void* const* d_in, const int* in_sizes, int n_in,
                              void* d_out, int out_size, void* d_ws, size_t ws_size,
                              hipStream_t stream) {
    const float* nodes = (const float*)d_in[0];   // [8,2048,16] f32
    const int*   T     = (const int*)d_in[1];     // [8] (JAX default x64-off -> int32)
    const int*   taus  = (const int*)d_in[2];     // [8]
    float*       out   = (float*)d_out;           // [8,2048,2048] f32

    const int total_tiles = NBATCH * TILES_PER_BATCH;        // 131072 waves
    dim3 grid(total_tiles / WAVES_PER_BLOCK);                // 16384 blocks
    dim3 block(256);                                         // 8 wave32 per block
    hipLaunchKernelGGL(SpatialRadiusEdge_kernel, grid, block, 0, stream,
                       nodes, T, taus, out);
}